// RNN_17102559773332
// MI455X (gfx1250) — compile-verified
//
#include <hip/hip_runtime.h>
#include <math.h>

// RNNCell forward for MI455X (gfx1250), fp32 via V_WMMA_F32_16X16X4_F32.
//
// Phase 1: xproj[t,b,:] = x[t,b,:] @ W_ih^T + (b_ih + b_hh), written into d_out.
//          Bandwidth-bound (~268 MB @ 23.3 TB/s); W_ih stays L2-hot.
// Phase 2: serial scan, batch-split across 4 independent workgroups.
//          W_hh^T lives in VGPRs (128/wave), h double-buffered in padded LDS,
//          xproj loads software-pipelined one timestep ahead so global-load
//          latency is off the serial critical path; outputs stored nontemporal.

typedef __attribute__((ext_vector_type(2))) float v2f;
typedef __attribute__((ext_vector_type(8))) float v8f;

#define T_STEPS 2048
#define BATCH   64
#define DIN     256
#define DH      256
#define LDS_STRIDE 260   // 256 + 4 pad -> conflict-free b64 A-fragment loads

__device__ __forceinline__ v8f wmma_f32(v2f a, v2f b, v8f c) {
  // (neg_a, A, neg_b, B, c_mod, C, reuse_a, reuse_b)
  return __builtin_amdgcn_wmma_f32_16x16x4_f32(
      false, a, false, b, (short)0, c, false, false);
}

// ---------------------------------------------------------------------------
// Phase 1: big GEMM. One wave per 16-row M tile, covering all 16 N tiles
// (full 256-wide row block) so every x element is loaded exactly once.
// 8192 M-tiles total -> 1024 blocks x 8 waves.
// ---------------------------------------------------------------------------
__global__ __launch_bounds__(256) void rnn_xproj_kernel(
    const float* __restrict__ x, const float* __restrict__ W_ih,
    const float* __restrict__ b_ih, const float* __restrict__ b_hh,
    float* __restrict__ out)
{
  const int lane  = threadIdx.x & 31;
  const int wave  = threadIdx.x >> 5;
  const int mtile = blockIdx.x * 8 + wave;   // 0..8191
  const int m     = lane & 15;
  const int kb    = (lane >> 4) * 2;         // K sub-offset per A/B layout

  // 16 accumulators (one per N tile), initialized with fused bias.
  v8f acc[16];
#pragma unroll
  for (int nt = 0; nt < 16; ++nt) {
    const int col = nt * 16 + m;
    const float bsum = b_ih[col] + b_hh[col];
#pragma unroll
    for (int r = 0; r < 8; ++r) acc[nt][r] = bsum;
  }

  // A fragment source: row (mtile*16 + m) of x, elements 4*kk+kb, +1 (contiguous pair)
  const float* arow = x + (size_t)(mtile * 16 + m) * DIN + kb;

#pragma unroll 4
  for (int kk = 0; kk < DIN / 4; ++kk) {
    const v2f a = *(const v2f*)(arow + 4 * kk);
#pragma unroll
    for (int nt = 0; nt < 16; ++nt) {
      // B[k][j] = W_ih[j][k]; lane holds col j = nt*16+m, k = 4*kk+kb (+1)
      const v2f b = *(const v2f*)(W_ih + (size_t)(nt * 16 + m) * DIN + kb + 4 * kk);
      acc[nt] = wmma_f32(a, b, acc[nt]);
    }
  }

  // C/D layout: VGPR r -> row r (lanes 0-15) / r+8 (lanes 16-31), col = lane&15
  const int rowoff = (lane >> 4) * 8;
  float* obase = out + (size_t)(mtile * 16) * DH;
#pragma unroll
  for (int nt = 0; nt < 16; ++nt) {
    const int col = nt * 16 + m;
#pragma unroll
    for (int r = 0; r < 8; ++r)
      obase[(size_t)(r + rowoff) * DH + col] = acc[nt][r];
  }
}

// ---------------------------------------------------------------------------
// Phase 2: recurrence. 4 workgroups (batch tiles of 16 rows), 16 waves each.
// Wave w owns output columns [16w, 16w+16); its W_hh^T fragments stay in VGPRs.
// h exchanged via double-buffered LDS; one barrier per timestep; xproj loads
// pipelined one step ahead of use.
// ---------------------------------------------------------------------------
__global__ __launch_bounds__(512) void rnn_scan_kernel(
    const float* __restrict__ hx, const float* __restrict__ W_hh,
    float* __restrict__ out)
{
  __shared__ float hbuf[2][16 * LDS_STRIDE];   // ~32.5 KB of the 320 KB WGP LDS

  const int lane  = threadIdx.x & 31;
  const int wave  = threadIdx.x >> 5;          // 0..15 -> N tile
  const int mbase = blockIdx.x * 16;           // batch row tile
  const int m     = lane & 15;
  const int kb    = (lane >> 4) * 2;

  // Resident W_hh^T fragments: B[k][j] = W_hh[j][k], j = wave*16 + m.
  v2f wv[64];
  {
    const float* wrow = W_hh + (size_t)(wave * 16 + m) * DH + kb;
#pragma unroll
    for (int kk = 0; kk < 64; ++kk) wv[kk] = *(const v2f*)(wrow + 4 * kk);
  }

  // Seed h(0) = hx into buffer 0.
  for (int idx = threadIdx.x; idx < 16 * DH; idx += 512) {
    const int mm = idx >> 8, k = idx & 255;
    hbuf[0][mm * LDS_STRIDE + k] = hx[(size_t)(mbase + mm) * DH + k];
  }
  __syncthreads();

  const int col    = wave * 16 + m;
  const int rowoff = (lane >> 4) * 8;

  // Software pipeline: prime xp_cur with xproj[0].
  float xp_cur[8], xp_next[8];
  {
    const float* orow0 = out + ((size_t)0 * BATCH + mbase) * DH;
#pragma unroll
    for (int r = 0; r < 8; ++r)
      xp_cur[r] = orow0[(size_t)(r + rowoff) * DH + col];
  }

  for (int t = 0; t < T_STEPS; ++t) {
    float* orow = out + ((size_t)t * BATCH + mbase) * DH;

    // Prefetch xproj[t+1]; its wait lands after this step's barrier, so the
    // global-load latency is fully hidden behind WMMA + tanh + exchange.
    if (t + 1 < T_STEPS) {
      const float* nrow = out + ((size_t)(t + 1) * BATCH + mbase) * DH;
#pragma unroll
      for (int r = 0; r < 8; ++r)
        xp_next[r] = nrow[(size_t)(r + rowoff) * DH + col];
    }

    const float* hprev = hbuf[t & 1];
    v8f acc;
#pragma unroll
    for (int r = 0; r < 8; ++r) acc[r] = 0.0f;

#pragma unroll
    for (int kk = 0; kk < 64; ++kk) {
      // A fragment: row m of h(t-1), elements 4*kk+kb, +1 (ds_load_b64, no conflicts)
      const v2f a = *(const v2f*)(hprev + m * LDS_STRIDE + 4 * kk + kb);
      acc = wmma_f32(a, wv[kk], acc);
    }

    float* hnext = hbuf[(t + 1) & 1];
#pragma unroll
    for (int r = 0; r < 8; ++r) {
      const float h = tanhf(acc[r] + xp_cur[r]);
      // outputs[t] is never re-read (next-step h comes from LDS): NT store
      // keeps the still-unread xproj lines resident in L2.
      __builtin_nontemporal_store(h, &orow[(size_t)(r + rowoff) * DH + col]);
      hnext[(r + rowoff) * LDS_STRIDE + col] = h;             // feed next step
    }
    __syncthreads();   // double buffer -> single barrier per step is sufficient

#pragma unroll
    for (int r = 0; r < 8; ++r) xp_cur[r] = xp_next[r];
  }

  // h_last lives right after outputs in d_out.
  float* hlast = out + (size_t)T_STEPS * BATCH * DH;
  const float* hfin = hbuf[T_STEPS & 1];
  for (int idx = threadIdx.x; idx < 16 * DH; idx += 512) {
    const int mm = idx >> 8, k = idx & 255;
    hlast[(size_t)(mbase + mm) * DH + k] = hfin[mm * LDS_STRIDE + k];
  }
}

// ---------------------------------------------------------------------------
extern "C" void kernel_launch(void* const* d_in, const int* in_sizes, int n_in,
                              void* d_out, int out_size, void* d_ws, size_t ws_size,
                              hipStream_t stream) {
  (void)in_sizes; (void)n_in; (void)out_size; (void)d_ws; (void)ws_size;
  const float* x    = (const float*)d_in[0];
  const float* hx   = (const float*)d_in[1];
  const float* W_ih = (const float*)d_in[2];
  const float* W_hh = (const float*)d_in[3];
  const float* b_ih = (const float*)d_in[4];
  const float* b_hh = (const float*)d_in[5];
  float* out = (float*)d_out;

  // Phase 1: 8192 M-tiles / 8 waves per block = 1024 blocks.
  rnn_xproj_kernel<<<dim3(1024), dim3(256), 0, stream>>>(x, W_ih, b_ih, b_hh, out);
  // Phase 2: 4 independent batch tiles.
  rnn_scan_kernel<<<dim3(BATCH / 16), dim3(512), 0, stream>>>(hx, W_hh, out);
}